// GNNRecommender_55499567399163
// MI455X (gfx1250) — compile-verified
//
#include <hip/hip_runtime.h>
#include <hip/hip_bf16.h>

#define DIM 128

typedef __attribute__((ext_vector_type(2))) float v2f;
typedef __attribute__((ext_vector_type(8))) float v8f;

#if defined(__has_builtin)
#if __has_builtin(__builtin_amdgcn_global_load_async_to_lds_b32) && \
    __has_builtin(__builtin_amdgcn_s_wait_asynccnt)
#define USE_ASYNC_LDS 1
#endif
#endif
#ifndef USE_ASYNC_LDS
#define USE_ASYNC_LDS 0
#endif

typedef __attribute__((address_space(1))) int gint_t;  // global addrspace
typedef __attribute__((address_space(3))) int lint_t;  // LDS addrspace

// ---------------------------------------------------------------------------
// Degree / normalization kernels
// ---------------------------------------------------------------------------
__global__ void k_init_deg(float* __restrict__ deg, int n) {
    int i = blockIdx.x * blockDim.x + threadIdx.x;
    if (i < n) deg[i] = 1.0f;                       // self-loop contribution
}

__global__ void k_edge_deg(const int* __restrict__ dst, float* __restrict__ deg, int E) {
    int e = blockIdx.x * blockDim.x + threadIdx.x;
    if (e < E) unsafeAtomicAdd(&deg[dst[e]], 1.0f); // global_atomic_add_f32
}

__global__ void k_dis(const float* __restrict__ deg, float* __restrict__ dis, int n) {
    int i = blockIdx.x * blockDim.x + threadIdx.x;
    if (i < n) dis[i] = rsqrtf(deg[i]);             // deg >= 1 always
}

// ---------------------------------------------------------------------------
// C[nRows,128] = A[nRows,128] @ W[128,128] (+ bias), f32 WMMA 16x16x4.
// W staged in LDS with K-pairs interleaved: lds[(k>>1)*256 + n*2 + (k&1)]
// so the per-lane B fragment (K', K'+1 at column n) is one 64-bit LDS load.
// Each wave computes a 16(M) x 128(N) strip: 8 accumulator tiles.
// ---------------------------------------------------------------------------
__global__ __launch_bounds__(256)
void k_gemm128_wmma(const float* __restrict__ A, const float* __restrict__ W,
                    const float* __restrict__ bias, float* __restrict__ C,
                    int nRows, int fuseBias)
{
    __shared__ float ldsW[64 * 256]; // 64 K-pairs * (128 cols * 2) = 64 KB

#if USE_ASYNC_LDS
    // Async global->LDS copy (scattered per-lane LDS destinations are legal:
    // dsaddr comes from a VGPR). Tracked by ASYNCcnt, waited before barrier.
    for (int i = threadIdx.x; i < DIM * DIM; i += 256) {
        int k = i >> 7, n = i & 127;
        __builtin_amdgcn_global_load_async_to_lds_b32(
            (gint_t*)(float*)(W + i),
            (lint_t*)&ldsW[(k >> 1) * 256 + n * 2 + (k & 1)],
            0, 0);
    }
    __builtin_amdgcn_s_wait_asynccnt(0);
    __syncthreads();
#else
    for (int i = threadIdx.x; i < DIM * DIM; i += 256) {
        int k = i >> 7, n = i & 127;
        ldsW[(k >> 1) * 256 + n * 2 + (k & 1)] = W[i];
    }
    __syncthreads();
#endif

    const int lane = threadIdx.x & 31;
    const int wave = threadIdx.x >> 5;
    const int half = lane >> 4;      // 0: lanes 0-15, 1: lanes 16-31
    const int l16  = lane & 15;
    const int m0   = (blockIdx.x * 8 + wave) * 16;
    if (m0 >= nRows) return;         // wave-uniform guard (EXEC stays all-ones)

    v8f acc[8];
    const v8f zero = {0.f, 0.f, 0.f, 0.f, 0.f, 0.f, 0.f, 0.f};
#pragma unroll
    for (int t = 0; t < 8; ++t) acc[t] = zero;

    const float2* Arow = (const float2*)(A + (size_t)(m0 + l16) * DIM);
    const float2* lds2 = (const float2*)ldsW;

#pragma unroll 4
    for (int ks = 0; ks < 32; ++ks) {
        // A fragment: lane half 0 holds K = 4ks+0,4ks+1 ; half 1 holds 4ks+2,4ks+3
        float2 af = Arow[ks * 2 + half];
        v2f a; a.x = af.x; a.y = af.y;
        const float2* brow = lds2 + (ks * 2 + half) * 128; // K-pair row
#pragma unroll
        for (int t = 0; t < 8; ++t) {
            float2 bf = brow[t * 16 + l16];
            v2f b; b.x = bf.x; b.y = bf.y;
            acc[t] = __builtin_amdgcn_wmma_f32_16x16x4_f32(
                false, a, false, b, (short)0, acc[t], false, false);
        }
    }

#pragma unroll
    for (int t = 0; t < 8; ++t) {
        int col = t * 16 + l16;
        float bv = fuseBias ? bias[col] : 0.0f;
#pragma unroll
        for (int r = 0; r < 8; ++r) {
            int row = m0 + r + 8 * half;   // C/D layout: VGPR r -> rows r, r+8
            C[(size_t)row * DIM + col] = acc[t][r] + bv;
        }
    }
}

// ---------------------------------------------------------------------------
// agg[i,:] = h[i,:] * dis[i]^2   (self-loop message, also initializes agg)
// float4 per thread: 32 float4 per node row.
// ---------------------------------------------------------------------------
__global__ void k_selfloop(const float* __restrict__ h, const float* __restrict__ dis,
                           float* __restrict__ agg, int total4) {
    int i = blockIdx.x * blockDim.x + threadIdx.x;
    if (i < total4) {
        float d  = dis[i >> 5];
        float dd = d * d;
        float4 v = ((const float4*)h)[i];
        v.x *= dd; v.y *= dd; v.z *= dd; v.w *= dd;
        ((float4*)agg)[i] = v;
    }
}

// ---------------------------------------------------------------------------
// Edge scatter-add: agg[dst,:] += h[src,:] * dis[src]*dis[dst]
// One wave per edge, float4 per lane: 512B coalesced gather (b128),
// 4 global_atomic_add_f32 per lane into L2-resident agg.
// ---------------------------------------------------------------------------
__global__ __launch_bounds__(256)
void k_scatter(const float* __restrict__ h, const float* __restrict__ dis,
               const int* __restrict__ src, const int* __restrict__ dst,
               float* __restrict__ agg, int E)
{
    int e = blockIdx.x * 8 + (threadIdx.x >> 5);
    if (e >= E) return;
    int lane = threadIdx.x & 31;
    int s = src[e], d = dst[e];
    float norm = dis[s] * dis[d];
    float4 hv = ((const float4*)(h + (size_t)s * DIM))[lane];
    float* ap = agg + (size_t)d * DIM + lane * 4;
    unsafeAtomicAdd(ap + 0, hv.x * norm);
    unsafeAtomicAdd(ap + 1, hv.y * norm);
    unsafeAtomicAdd(ap + 2, hv.z * norm);
    unsafeAtomicAdd(ap + 3, hv.w * norm);
}

// ---------------------------------------------------------------------------
// out = relu(agg + b), float4 per thread
// ---------------------------------------------------------------------------
__global__ void k_bias_relu(const float* __restrict__ agg, const float* __restrict__ b,
                            float* __restrict__ out, int total4) {
    int i = blockIdx.x * blockDim.x + threadIdx.x;
    if (i < total4) {
        float4 v  = ((const float4*)agg)[i];
        float4 bv = ((const float4*)b)[i & 31];
        v.x += bv.x; v.y += bv.y; v.z += bv.z; v.w += bv.w;
        v.x = v.x > 0.f ? v.x : 0.f;
        v.y = v.y > 0.f ? v.y : 0.f;
        v.z = v.z > 0.f ? v.z : 0.f;
        v.w = v.w > 0.f ? v.w : 0.f;
        ((float4*)out)[i] = v;
    }
}

// ---------------------------------------------------------------------------
extern "C" void kernel_launch(void* const* d_in, const int* in_sizes, int n_in,
                              void* d_out, int out_size, void* d_ws, size_t ws_size,
                              hipStream_t stream)
{
    const float* x_user = (const float*)d_in[0];
    const float* x_item = (const float*)d_in[1];
    const int*   e_ui   = (const int*)d_in[2];
    const int*   e_iu   = (const int*)d_in[3];
    const float* W_u2i  = (const float*)d_in[4];
    const float* b_u2i  = (const float*)d_in[5];
    const float* W_i2u  = (const float*)d_in[6];
    const float* b_i2u  = (const float*)d_in[7];
    const float* W_proj = (const float*)d_in[8];
    const float* b_proj = (const float*)d_in[9];

    const int n    = in_sizes[0] / DIM;   // 100000 nodes
    const int E_ui = in_sizes[2] / 2;
    const int E_iu = in_sizes[3] / 2;

    float* out = (float*)d_out;
    float* h   = (float*)d_ws;                    // [n,128]
    float* agg = h + (size_t)n * DIM;             // [n,128]
    float* deg = agg + (size_t)n * DIM;           // [n]
    float* dis = deg + n;                         // [n]

    const int total4 = n * DIM / 4;
    const int gElem  = (total4 + 255) / 256;
    const int gN     = (n + 255) / 256;
    const int gGemm  = ((n + 15) / 16 + 7) / 8;

    struct Branch { const float* x; const int* edges; const float* Wc;
                    const float* bc; float* outPart; int E; };
    Branch br[2] = {
        { x_item, e_ui, W_u2i, b_u2i, out,                   E_ui },  // item_emb
        { x_user, e_iu, W_i2u, b_i2u, out + (size_t)n * DIM, E_iu },  // user_emb
    };

    for (int i = 0; i < 2; ++i) {
        const int* src = br[i].edges;
        const int* dst = br[i].edges + br[i].E;

        k_init_deg<<<gN, 256, 0, stream>>>(deg, n);
        k_edge_deg<<<(br[i].E + 255) / 256, 256, 0, stream>>>(dst, deg, br[i].E);
        k_dis<<<gN, 256, 0, stream>>>(deg, dis, n);

        // h = x @ W_conv
        k_gemm128_wmma<<<gGemm, 256, 0, stream>>>(br[i].x, br[i].Wc, nullptr, h, n, 0);

        // aggregation: self-loops then edge scatter-add
        k_selfloop<<<gElem, 256, 0, stream>>>(h, dis, agg, total4);
        k_scatter<<<(br[i].E + 7) / 8, 256, 0, stream>>>(h, dis, src, dst, agg, br[i].E);

        // emb = relu(agg + b_conv), written directly into the output slot
        k_bias_relu<<<gElem, 256, 0, stream>>>(agg, br[i].bc, br[i].outPart, total4);

        // out = emb @ W_proj + b_proj (in place: each wave reads only rows it writes)
        k_gemm128_wmma<<<gGemm, 256, 0, stream>>>(br[i].outPart, W_proj, b_proj,
                                                  br[i].outPart, n, 1);
    }
}